// kWTA_34050500722819
// MI455X (gfx1250) — compile-verified
//
#include <hip/hip_runtime.h>

// kWTA: per-sample exact top-k threshold via 16-bit radix select in LDS,
// then mask. Shapes fixed by reference: x is (128, 256, 32, 32) f32.
#define CHW        262144            // 256*32*32 elements per sample
#define NV4        (CHW / 4)         // 65536 float4 per sample
#define THREADS    1024              // 32 wave32 waves -> fills a WGP
#define ITER       (NV4 / THREADS)   // 64 float4 per thread per pass
#define KSEL       26214u            // int(0.1 * 262144)
#define NBINS      65536
#define NCHUNK     THREADS
#define BPC        (NBINS / NCHUNK)  // 64 bins per thread chunk

// LDS budget: 65536 hist + 1024 chunk sums + 1024 suffix + 4 control words
// = 270,352 bytes. Legal only because CDNA5 WGP has 320 KB LDS.
#define SMEM_WORDS (NBINS + 2 * NCHUNK + 4)

// native vector type: accepted by __builtin_nontemporal_store, lowers to b128
typedef float v4f __attribute__((ext_vector_type(4)));

__device__ __forceinline__ unsigned int f2key(float f) {
  unsigned int u = __float_as_uint(f);
  // order-preserving bijection: larger float <=> larger unsigned key
  return (u & 0x80000000u) ? ~u : (u | 0x80000000u);
}

__device__ __forceinline__ float key2f(unsigned int k) {
  unsigned int u = (k & 0x80000000u) ? (k & 0x7FFFFFFFu) : ~k;
  return __uint_as_float(u);
}

__global__ __launch_bounds__(THREADS) void kwta_kernel(
    const float* __restrict__ x, float* __restrict__ out) {
  extern __shared__ unsigned int smem[];
  unsigned int* hist = smem;                       // [NBINS]
  unsigned int* csum = smem + NBINS;               // [NCHUNK] chunk sums
  unsigned int* cabv = smem + NBINS + NCHUNK;      // [NCHUNK] suffix-above
  unsigned int* sel  = smem + NBINS + 2 * NCHUNK;  // [0]=bin [1]=k' [2]=bits

  const int t = threadIdx.x;
  const v4f* __restrict__ xin = (const v4f*)(x + (size_t)blockIdx.x * CHW);
  v4f* __restrict__ xout = (v4f*)(out + (size_t)blockIdx.x * CHW);

  // ---------------- Pass 1: histogram of top 16 key bits ----------------
  for (int i = t; i < NBINS; i += THREADS) hist[i] = 0u;
  __syncthreads();

  #pragma unroll 4
  for (int i = 0; i < ITER; ++i) {
    // prefetch a few tiles ahead on the cold HBM streaming pass
    __builtin_prefetch(&xin[t + (i + 8) * THREADS], 0, 0);
    v4f v = xin[t + i * THREADS];
    atomicAdd(&hist[f2key(v.x) >> 16], 1u);
    atomicAdd(&hist[f2key(v.y) >> 16], 1u);
    atomicAdd(&hist[f2key(v.z) >> 16], 1u);
    atomicAdd(&hist[f2key(v.w) >> 16], 1u);
  }
  __syncthreads();

  // ---- Select the top-16 bin containing the KSEL-th largest element ----
  {
    unsigned int s = 0;
    #pragma unroll
    for (int i = 0; i < BPC; ++i) s += hist[t * BPC + i];
    csum[t] = s;
  }
  __syncthreads();
  if (t == 0) {
    unsigned int running = 0;
    for (int c = NCHUNK - 1; c >= 0; --c) {  // suffix over higher chunks
      cabv[c] = running;
      running += csum[c];
    }
  }
  __syncthreads();
  {
    unsigned int above = cabv[t];
    if (above < KSEL && above + csum[t] >= KSEL) {  // crossing in my chunk
      unsigned int run = above;
      for (int i = BPC - 1; i >= 0; --i) {
        unsigned int bin = (unsigned)(t * BPC + i);
        unsigned int c = hist[bin];
        if (run + c >= KSEL) {
          sel[0] = bin;
          sel[1] = KSEL - run;  // residual rank within this bin (>=1)
          break;
        }
        run += c;
      }
    }
  }
  __syncthreads();
  const unsigned int topbin = sel[0];
  const unsigned int k2 = sel[1];
  __syncthreads();

  // ---------------- Pass 2: low 16 bits within the top bin ----------------
  for (int i = t; i < NBINS; i += THREADS) hist[i] = 0u;
  __syncthreads();

  #pragma unroll 4
  for (int i = 0; i < ITER; ++i) {
    v4f v = xin[t + i * THREADS];  // L2-resident re-read
    unsigned int kx = f2key(v.x);
    unsigned int ky = f2key(v.y);
    unsigned int kz = f2key(v.z);
    unsigned int kw = f2key(v.w);
    if ((kx >> 16) == topbin) atomicAdd(&hist[kx & 0xFFFFu], 1u);
    if ((ky >> 16) == topbin) atomicAdd(&hist[ky & 0xFFFFu], 1u);
    if ((kz >> 16) == topbin) atomicAdd(&hist[kz & 0xFFFFu], 1u);
    if ((kw >> 16) == topbin) atomicAdd(&hist[kw & 0xFFFFu], 1u);
  }
  __syncthreads();

  // ---- Select the low-16 bin with rank k2 -> exact threshold ----
  {
    unsigned int s = 0;
    #pragma unroll
    for (int i = 0; i < BPC; ++i) s += hist[t * BPC + i];
    csum[t] = s;
  }
  __syncthreads();
  if (t == 0) {
    unsigned int running = 0;
    for (int c = NCHUNK - 1; c >= 0; --c) {
      cabv[c] = running;
      running += csum[c];
    }
  }
  __syncthreads();
  {
    unsigned int above = cabv[t];
    if (above < k2 && above + csum[t] >= k2) {
      unsigned int run = above;
      for (int i = BPC - 1; i >= 0; --i) {
        unsigned int bin = (unsigned)(t * BPC + i);
        unsigned int c = hist[bin];
        if (run + c >= k2) {
          sel[2] = __float_as_uint(key2f((topbin << 16) | bin));
          break;
        }
        run += c;
      }
    }
  }
  __syncthreads();
  const float thresh = __uint_as_float(sel[2]);

  // ---------------- Pass 3: apply mask, NT store (write-once) -------------
  #pragma unroll 4
  for (int i = 0; i < ITER; ++i) {
    v4f v = xin[t + i * THREADS];  // L2-resident re-read
    v4f o;
    o.x = (v.x >= thresh) ? v.x : 0.0f;
    o.y = (v.y >= thresh) ? v.y : 0.0f;
    o.z = (v.z >= thresh) ? v.z : 0.0f;
    o.w = (v.w >= thresh) ? v.w : 0.0f;
    __builtin_nontemporal_store(o, &xout[t + i * THREADS]);
  }
}

extern "C" void kernel_launch(void* const* d_in, const int* in_sizes, int n_in,
                              void* d_out, int out_size, void* d_ws,
                              size_t ws_size, hipStream_t stream) {
  (void)n_in; (void)d_ws; (void)ws_size; (void)out_size;
  const float* x = (const float*)d_in[0];
  float* out = (float*)d_out;
  const int batches = in_sizes[0] / CHW;  // 128 for the reference shapes
  const size_t smem_bytes = SMEM_WORDS * sizeof(unsigned int);  // ~264 KB
  kwta_kernel<<<batches, THREADS, smem_bytes, stream>>>(x, out);
}